// SNNLayer_47536698032233
// MI455X (gfx1250) — compile-verified
//
#include <hip/hip_runtime.h>
#include <hip/hip_bf16.h>
#include <stdint.h>

// SNN spike-time layer for MI455X (gfx1250, wave32).
//
// Kernel 1: per-batch-row bitonic argsort in LDS (2048-slot, key packs
//           (value_bits<<32)|index -> stable, order-preserving for v>=0).
// Kernel 2: blocked scan over the sorted axis. Within-block (16-wide)
//           cumulative sums are done as T(16x16 lower-tri ones) @ W(16x16)
//           via V_WMMA_F32_16X16X4_F32 (full f32 precision), one wave per
//           (batch, 16-column) tile; carry propagated via lane shuffle.

#define MAX_SPIKE_TIME 100000.0f
#define B_SZ    64
#define IN_SZ   1024
#define NX      1025     // 1024 inputs + ones column
#define OUT_SZ  512
#define SORT_N  2048
#define XS_PAD  1056     // padded sorted-value array (need index up to 1040)
#define IDX_PAD 1040     // padded index array (k up to 1039)

typedef float v2f __attribute__((ext_vector_type(2)));
typedef float v8f __attribute__((ext_vector_type(8)));

// ---------------------------------------------------------------------------
// Kernel 1: x = in * exp(relu(delay)), append 1.0, stable ascending argsort.
// One 1024-thread workgroup per batch row; bitonic sort of 2048 packed keys.
// ---------------------------------------------------------------------------
__global__ __launch_bounds__(1024) void snn_sort_kernel(
    const float* __restrict__ layer_in, const float* __restrict__ delay,
    float* __restrict__ xs_ws, int* __restrict__ idx_ws)
{
  __shared__ uint64_t keys[SORT_N];
  const int b = blockIdx.x;
  const int t = threadIdx.x;

  for (int i = t; i < SORT_N; i += 1024) {
    uint64_t key;
    if (i < IN_SZ) {
      float d = delay[i];
      float v = layer_in[b * IN_SZ + i] * expf(fmaxf(d, 0.0f));
      key = (((uint64_t)__float_as_uint(v)) << 32) | (uint32_t)i;
    } else if (i == IN_SZ) {
      key = (((uint64_t)__float_as_uint(1.0f)) << 32) | (uint32_t)i;
    } else {
      key = ~0ull;  // sorts past the end
    }
    keys[i] = key;
  }
  __syncthreads();

  for (int k = 2; k <= SORT_N; k <<= 1) {
    for (int j = k >> 1; j > 0; j >>= 1) {
      for (int i = t; i < SORT_N; i += 1024) {
        int ixj = i ^ j;
        if (ixj > i) {  // owner of the pair's lower index does the exchange
          uint64_t a = keys[i], c = keys[ixj];
          bool asc = ((i & k) == 0);
          if ((a > c) == asc) { keys[i] = c; keys[ixj] = a; }
        }
      }
      __syncthreads();
    }
  }

  for (int i = t; i < XS_PAD; i += 1024) {
    float v; int id;
    if (i < NX) {
      uint64_t key = keys[i];
      v  = __uint_as_float((uint32_t)(key >> 32));
      id = (int)(uint32_t)(key & 0xFFFFFFFFu);
    } else {
      v = MAX_SPIKE_TIME;  // padded sorted values -> candidates become MAX
      id = 0;              // padded gather index (weight masked to 0 anyway)
    }
    xs_ws[b * XS_PAD + i] = v;
    if (i < IDX_PAD) idx_ws[b * IDX_PAD + i] = id;
  }
}

// ---------------------------------------------------------------------------
// Kernel 2: blocked cumulative scan + candidate min via f32 WMMA.
// One 1024-thread workgroup (32 waves) per batch row; wave w owns columns
// [16w, 16w+16). Per 16-row k-block: C += T @ W with 4x V_WMMA_F32_16X16X4_F32
// per cumsum (w_cum and wi_cum). Carry = cumsum row 15, shuffled from lanes
// 16..31 VGPR7 and splatted into all 8 accumulator regs for the next block.
//
// ISA layouts (CDNA5 ISA):
//   A 16x4 f32: M = lane&15, K = 2*(lane>>4) + reg          (documented)
//   B 4x16 f32: N = lane&15, K = 2*(lane>>4) + reg          (assumed mirror)
//   C 16x16 f32: N = lane&15, M = reg + 8*(lane>>4)         (documented)
// ---------------------------------------------------------------------------
__global__ __launch_bounds__(1024) void snn_scan_wmma_kernel(
    const float* __restrict__ weight, const float* __restrict__ xs_ws,
    const int* __restrict__ idx_ws, float* __restrict__ out)
{
  __shared__ float s_xs[XS_PAD];
  __shared__ int   s_idx[IDX_PAD];
  const int b = blockIdx.x;
  const int t = threadIdx.x;

  for (int i = t; i < XS_PAD; i += 1024)  s_xs[i]  = xs_ws[b * XS_PAD + i];
  for (int i = t; i < IDX_PAD; i += 1024) s_idx[i] = idx_ws[b * IDX_PAD + i];
  __syncthreads();

  const int wave = t >> 5;          // 0..31 -> j-tile
  const int lane = t & 31;
  const int n    = lane & 15;       // column within tile (also A-matrix row M)
  const int half = lane >> 4;       // lane half selects K/M sub-range
  const int j    = wave * 16 + n;   // global output column

  v8f cw  = {0.f,0.f,0.f,0.f,0.f,0.f,0.f,0.f};   // w_cum accumulator tile
  v8f cwi = {0.f,0.f,0.f,0.f,0.f,0.f,0.f,0.f};   // wi_cum accumulator tile
  float vmin = MAX_SPIKE_TIME;

  for (int kb = 0; kb < IDX_PAD; kb += 16) {
#pragma unroll
    for (int c = 0; c < 4; ++c) {
      const int kk = c * 4 + half * 2;      // block-local K for this lane's regs
      // A operand: lower-triangular ones  T[m, kk] = (kk <= m)
      v2f a;
      a.x = (kk     <= n) ? 1.0f : 0.0f;
      a.y = (kk + 1 <= n) ? 1.0f : 0.0f;
      // B operand: gathered weight rows in sorted order (0 past k=1024)
      const int k0 = kb + kk, k1 = k0 + 1;  // both < IDX_PAD by construction
      const float m0 = (k0 <= IN_SZ) ? 1.0f : 0.0f;
      const float m1 = (k1 <= IN_SZ) ? 1.0f : 0.0f;
      const float w0 = weight[s_idx[k0] * OUT_SZ + j] * m0;
      const float w1 = weight[s_idx[k1] * OUT_SZ + j] * m1;
      v2f bw  = { w0, w1 };
      v2f bwi = { w0 * s_xs[k0], w1 * s_xs[k1] };   // ws * xs (f32, as reference)
      cw  = __builtin_amdgcn_wmma_f32_16x16x4_f32(false, a, false, bw,
                                                  (short)0, cw,  false, false);
      cwi = __builtin_amdgcn_wmma_f32_16x16x4_f32(false, a, false, bwi,
                                                  (short)0, cwi, false, false);
    }

    // Candidate evaluation for the 16 sorted positions in this block.
#pragma unroll
    for (int r = 0; r < 8; ++r) {
      const int m = r + half * 8;
      const int k = kb + m;
      const float wc  = cw[r];
      const float wic = cwi[r];
      const float xk  = s_xs[k];
      const float xk1 = s_xs[k + 1];
      const float out_all = wic / fmaxf(wc - 1.0f, 1e-10f);
      float cand = (wc < 1.0f)   ? MAX_SPIKE_TIME : out_all;
      cand       = (cand < xk)   ? MAX_SPIKE_TIME : cand;
      cand       = (cand > xk1)  ? MAX_SPIKE_TIME : cand;
      vmin = fminf(vmin, cand);
    }

    // Carry for next block = cumsum row M=15 (VGPR7, lanes 16..31), splat.
    const float carry_w  = __shfl(cw[7],  16 + n, 32);
    const float carry_wi = __shfl(cwi[7], 16 + n, 32);
#pragma unroll
    for (int r = 0; r < 8; ++r) { cw[r] = carry_w; cwi[r] = carry_wi; }
  }

  // Combine the two lane halves (rows m and m+8 minima, same column j).
  vmin = fminf(vmin, __shfl(vmin, lane ^ 16, 32));
  if (half == 0) out[b * OUT_SZ + j] = vmin;
}

// ---------------------------------------------------------------------------
extern "C" void kernel_launch(void* const* d_in, const int* in_sizes, int n_in,
                              void* d_out, int out_size, void* d_ws, size_t ws_size,
                              hipStream_t stream)
{
  (void)in_sizes; (void)n_in; (void)out_size; (void)ws_size;
  const float* layer_in = (const float*)d_in[0];   // (64, 1024) f32
  const float* weight   = (const float*)d_in[1];   // (1025, 512) f32
  const float* delay    = (const float*)d_in[2];   // (1024,) f32
  float* out = (float*)d_out;                      // (64, 512) f32

  float* xs_ws  = (float*)d_ws;                                     // 64*1056 f32
  int*   idx_ws = (int*)((char*)d_ws + B_SZ * XS_PAD * sizeof(float)); // 64*1040 i32

  snn_sort_kernel<<<B_SZ, 1024, 0, stream>>>(layer_in, delay, xs_ws, idx_ws);
  snn_scan_wmma_kernel<<<B_SZ, 1024, 0, stream>>>(weight, xs_ws, idx_ws, out);
}